// CurveEval3_80779744903774
// MI455X (gfx1250) — compile-verified
//
#include <hip/hip_runtime.h>

typedef __attribute__((ext_vector_type(2))) float v2f;
typedef __attribute__((ext_vector_type(8))) float v8f;

#define M_CP   128
#define P_DEG  3
#define OUTD   256
#define DIM    3
#define KNOTS  (M_CP + P_DEG + 1)   // 132
#define CROWS  (M_CP + 4)           // 132 rows; rows >= 128 are zero pad
#define NT     256

// Async global -> LDS (GLOBAL_LOAD_ASYNC_TO_LDS_B32, ASYNCcnt).
// Low 32 bits of a generic LDS pointer are the LDS byte offset (ISA 10.2).
__device__ __forceinline__ void async_g2l_b32(const float* __restrict__ g, float* l) {
    asm volatile("global_load_async_to_lds_b32 %0, %1, off"
                 :
                 : "v"((uint32_t)(uintptr_t)l),
                   "v"((uint64_t)(uintptr_t)g)
                 : "memory");
}

// Async LDS -> global (GLOBAL_STORE_ASYNC_FROM_LDS_B128, ASYNCcnt).
__device__ __forceinline__ void async_l2g_b128(float* g, const float* l) {
    asm volatile("global_store_async_from_lds_b128 %0, %1, off"
                 :
                 : "v"((uint64_t)(uintptr_t)g),
                   "v"((uint32_t)(uintptr_t)l)
                 : "memory");
}

__global__ __launch_bounds__(NT) void curve_eval3_wmma(
    const float* __restrict__ ctrl,    // [B, M_CP, DIM]
    const float* __restrict__ knots,   // [B, KNOTS]
    float* __restrict__ out)           // [B, OUTD, DIM]
{
    __shared__ float Ulds[KNOTS];
    __shared__ float Clds[CROWS * 16];             // zero-padded B operand [k][n]
    __shared__ float NiLds[OUTD * 4];              // basis values per eval point
    __shared__ int   SpanLds[OUTD];                // knot span per eval point
    __shared__ __align__(16) float Olds[OUTD * DIM];  // output staging (16B aligned)

    const int b   = blockIdx.x;
    const int tid = threadIdx.x;

    // ---------- stage per-curve data via async-to-LDS ----------
    if (tid < KNOTS)
        async_g2l_b32(knots + b * KNOTS + tid, &Ulds[tid]);
    for (int idx = tid; idx < M_CP * DIM; idx += NT)
        async_g2l_b32(ctrl + b * (M_CP * DIM) + idx,
                      &Clds[(idx / DIM) * 16 + (idx % DIM)]);
    // zero the padding cells (disjoint addresses -> no ordering hazard)
    for (int i = tid; i < CROWS * 16; i += NT) {
        const int m = i >> 4, d = i & 15;
        if (d >= DIM || m >= M_CP) Clds[i] = 0.0f;
    }
    asm volatile("s_wait_asynccnt 0x0" ::: "memory");
    __syncthreads();

    // ---------- phase 1: one eval point per thread ----------
    {
        const int   t = tid;
        const float u = 1e-5f + (float)t * ((1.0f - 2e-5f) / 255.0f);

        // d_j = u - U[p+j] is non-increasing in j, so {d > 1e-8} is a prefix.
        // Reference masked-argmin == first index of the value-group of the
        // last prefix element. Binary search (7 steps) + duplicate walk-back.
        int lo = 0, hi = KNOTS - 1 - P_DEG;         // cond(0)=true, cond(128)=false
        while (hi - lo > 1) {
            const int mid = (lo + hi) >> 1;
            if (u - Ulds[P_DEG + mid] > 1e-8f) lo = mid; else hi = mid;
        }
        int jL = lo;
        const float V = Ulds[P_DEG + jL];
        while (jL > 0 && Ulds[P_DEG + jL - 1] == V) --jL;   // first occurrence
        const int uspan = jL + P_DEG;                       // in [3, 127]

        // Cox–de Boor, degree 3, fully unrolled in registers
        float Ni[P_DEG + 1] = {1.0f, 0.0f, 0.0f, 0.0f};
        #pragma unroll
        for (int k = 1; k <= P_DEG; ++k) {
            float saved = 0.0f;
            #pragma unroll
            for (int r = 0; r < k; ++r) {
                float U1   = Ulds[uspan + r + 1];
                float U2   = Ulds[uspan + 1 - k + r];
                float dU   = (U1 - u) + (u - U2);
                float dU_  = (dU == 0.0f) ? 1e-4f : dU;
                float temp = Ni[r] / dU_;
                temp       = (dU == 0.0f) ? 1e-4f : temp;
                Ni[r] = saved + (U1 - u) * temp;
                saved = (u - U2) * temp;
            }
            Ni[k] = saved;
        }

        SpanLds[t] = uspan;
        #pragma unroll
        for (int k = 0; k <= P_DEG; ++k) NiLds[t * 4 + k] = Ni[k];
    }
    __syncthreads();

    // ---------- phase 2: banded GEMM tiles via V_WMMA_F32_16X16X4_F32 ----------
    const int lane     = tid & 31;
    const int wave     = tid >> 5;
    const int col      = lane & 15;     // A row (t in tile) / B,D column
    const int half     = lane >> 4;
    const int klaneoff = half * 2;      // A/B element i holds K = i + klaneoff
    const int rowoff   = half * 8;      // D element j holds M = j + rowoff

    for (int tile = wave; tile < OUTD / 16; tile += 8) {
        const int t = tile * 16 + col;

        // spans are monotone in t: tile window from first/last row, made
        // wave-uniform (SGPR) so the K-loop is scalar and EXEC stays all-ones
        const int klo = __builtin_amdgcn_readfirstlane(SpanLds[tile * 16]) - P_DEG;
        const int khi = __builtin_amdgcn_readfirstlane(SpanLds[tile * 16 + 15]);
        const int span = SpanLds[t];

        v8f c = {};
        for (int kk = klo; kk <= khi; kk += 4) {
            v2f a, bm;
            #pragma unroll
            for (int i = 0; i < 2; ++i) {
                const int kg = kk + i + klaneoff;       // global ctrl index <= 130
                const int r  = kg - span + P_DEG;       // valid band: 0..3
                const int rc = r & 3;                   // wrapped, masked below
                float av = NiLds[t * 4 + rc];           // unconditional ds_load
                av = (r == rc) ? av : 0.0f;             // single cndmask
                a[i]  = av;
                bm[i] = Clds[kg * 16 + col];            // unconditional ds_load
            }
            c = __builtin_amdgcn_wmma_f32_16x16x4_f32(
                    false, a, false, bm, (short)0, c, false, false);
        }

        if (col < DIM) {
            #pragma unroll
            for (int j = 0; j < 8; ++j)
                Olds[(tile * 16 + j + rowoff) * DIM + col] = c[j];
        }
    }
    __syncthreads();

    // ---------- phase 3: async LDS -> global, 192 x 16B per curve ----------
    for (int i = tid; i < (OUTD * DIM) / 4; i += NT)
        async_l2g_b128(out + b * (OUTD * DIM) + i * 4, &Olds[i * 4]);
    asm volatile("s_wait_asynccnt 0x0" ::: "memory");
}

extern "C" void kernel_launch(void* const* d_in, const int* in_sizes, int n_in,
                              void* d_out, int out_size, void* d_ws, size_t ws_size,
                              hipStream_t stream) {
    const float* ctrl  = (const float*)d_in[0];   // (4096, 128, 3) f32
    const float* knots = (const float*)d_in[1];   // (4096, 132)    f32
    float* out = (float*)d_out;                   // (4096, 256, 3) f32

    const int B = in_sizes[1] / KNOTS;            // 4096
    curve_eval3_wmma<<<B, NT, 0, stream>>>(ctrl, knots, out);
}